// RouteGNN_8701603742460
// MI455X (gfx1250) — compile-verified
//
#include <hip/hip_runtime.h>

typedef float v2f __attribute__((ext_vector_type(2)));
typedef float v8f __attribute__((ext_vector_type(8)));

#define CH     128     // IN_CH == HID == 128
#define NCLS   32
#define NGRAPH 256

// ---------------------------------------------------------------------------
// GEMM: Y[N,128] = X[N,128] @ W[128,128] using V_WMMA_F32_16X16X4_F32.
// Block = 256 threads (8 waves). Block owns a 16-row tile of X staged in LDS
// (row stride 132 floats to avoid 16-way LDS bank conflicts on the strided
// A-fragment reads). Wave w owns output columns [16w, 16w+16). K walks 0..127
// in steps of 4 -> 32 WMMA ops per wave.
//
// Fragment layouts (ISA 7.12.2):
//   A 16x4 : lanes 0-15 (rows 0-15) hold K=0,1 in {v.x,v.y}; lanes 16-31 hold K=2,3
//   B 4x16 : lane r = col n0+r; lanes 0-15 hold K=k0,k0+1; lanes 16-31 hold K=k0+2,k0+3
//   C/D    : VGPR v, lane r, half h -> (row m0 + v + 8h, col n0 + r)
// ---------------------------------------------------------------------------
__global__ __launch_bounds__(256)
void gemm_wmma_f32(const float* __restrict__ X, const float* __restrict__ W,
                   float* __restrict__ Y, int n_rows) {
  __shared__ float lds[16 * 132];
  const int m0 = blockIdx.x * 16;
  const int t  = threadIdx.x;

  // Cooperative coalesced load of the 16x128 tile (512 float4, 2 per thread).
  for (int idx = t; idx < 512; idx += 256) {
    const int row = idx >> 5;        // 0..15
    const int c4  = idx & 31;        // 0..31 float4 within the row
    const int gr  = m0 + row;
    float4 v = make_float4(0.f, 0.f, 0.f, 0.f);
    if (gr < n_rows) v = ((const float4*)(X + (size_t)gr * CH))[c4];
    *(float4*)&lds[row * 132 + c4 * 4] = v;
  }
  __syncthreads();

  const int wave = t >> 5;
  const int lane = t & 31;
  const int half = lane >> 4;        // 0 or 1
  const int r    = lane & 15;
  const int n0   = wave * 16;

  v8f acc = {};
  #pragma unroll
  for (int k0 = 0; k0 < CH; k0 += 4) {
    const int ka = k0 + 2 * half;
    v2f a, b;
    a.x = lds[r * 132 + ka];
    a.y = lds[r * 132 + ka + 1];
    b.x = W[(size_t)ka       * CH + n0 + r];
    b.y = W[(size_t)(ka + 1) * CH + n0 + r];
    acc = __builtin_amdgcn_wmma_f32_16x16x4_f32(
        /*neg_a=*/false, a, /*neg_b=*/false, b,
        /*c_mod=*/(short)0, acc, /*reuse_a=*/false, /*reuse_b=*/false);
  }

  #pragma unroll
  for (int v = 0; v < 8; ++v) {
    const int row = m0 + v + 8 * half;
    if (row < n_rows) Y[(size_t)row * CH + n0 + r] = acc[v];
  }
}

// ---------------------------------------------------------------------------
// Degree / normalization
// ---------------------------------------------------------------------------
__global__ void deg_init(float* __restrict__ deg, int n) {
  int i = blockIdx.x * blockDim.x + threadIdx.x;
  if (i < n) deg[i] = 1.0f;                      // self-loop contribution
}

__global__ void deg_count(const int* __restrict__ dst, float* __restrict__ deg, int E) {
  int i = blockIdx.x * blockDim.x + threadIdx.x;
  if (i < E) atomicAdd(&deg[dst[i]], 1.0f);
}

__global__ void deg_rsqrt(const float* __restrict__ deg, float* __restrict__ dinv, int n) {
  int i = blockIdx.x * blockDim.x + threadIdx.x;
  if (i < n) {
    float d = deg[i];
    dinv[i] = (d > 0.f) ? rsqrtf(d) : 0.f;
  }
}

// agg[i,:] = xw[i,:] * dinv[i]^2   (self-loop message; also zero-initializes agg)
__global__ void self_loop_init(const float* __restrict__ xw, const float* __restrict__ dinv,
                               float* __restrict__ agg, int n) {
  int idx = blockIdx.x * blockDim.x + threadIdx.x;   // over n * 32 float4
  if (idx >= n * 32) return;
  const int node = idx >> 5;
  const int c4   = idx & 31;
  float s = dinv[node];
  s = s * s;
  float4 v = ((const float4*)(xw + (size_t)node * CH))[c4];
  v.x *= s; v.y *= s; v.z *= s; v.w *= s;
  ((float4*)(agg + (size_t)node * CH))[c4] = v;
}

// One wave per edge: coalesced 512B gather of xw[src], scale, atomic scatter to agg[dst].
__global__ __launch_bounds__(256)
void edge_scatter(const int* __restrict__ src, const int* __restrict__ dst,
                  const float* __restrict__ dinv, const float* __restrict__ xw,
                  float* __restrict__ agg, int E) {
  const int e = blockIdx.x * 8 + (threadIdx.x >> 5);
  if (e >= E) return;
  const int lane = threadIdx.x & 31;
  const int s = src[e];
  const int d = dst[e];
  const float nrm = dinv[s] * dinv[d];
  float4 v = ((const float4*)(xw + (size_t)s * CH))[lane];
  float* o = agg + (size_t)d * CH + lane * 4;
  atomicAdd(o + 0, v.x * nrm);
  atomicAdd(o + 1, v.y * nrm);
  atomicAdd(o + 2, v.z * nrm);
  atomicAdd(o + 3, v.w * nrm);
}

__global__ void bias_relu(float* __restrict__ h, const float* __restrict__ b, size_t total) {
  size_t i = (size_t)blockIdx.x * blockDim.x + threadIdx.x;
  if (i < total) {
    float v = h[i] + b[i & (CH - 1)];
    h[i] = v > 0.f ? v : 0.f;
  }
}

// ---------------------------------------------------------------------------
// Global mean pool + FC
// ---------------------------------------------------------------------------
__global__ void pool_zero(float* __restrict__ pool, float* __restrict__ cnt) {
  int i = blockIdx.x * blockDim.x + threadIdx.x;
  if (i < NGRAPH * CH) pool[i] = 0.f;
  if (i < NGRAPH) cnt[i] = 0.f;
}

__global__ void pool_count(const int* __restrict__ batch, float* __restrict__ cnt, int n) {
  int i = blockIdx.x * blockDim.x + threadIdx.x;
  if (i < n) atomicAdd(&cnt[batch[i]], 1.0f);
}

__global__ void pool_sum(const float* __restrict__ h, const int* __restrict__ batch,
                         float* __restrict__ pool, size_t total) {
  size_t i = (size_t)blockIdx.x * blockDim.x + threadIdx.x;
  if (i >= total) return;
  const int node = (int)(i >> 7);       // / 128
  const int c    = (int)(i & (CH - 1));
  atomicAdd(&pool[(size_t)batch[node] * CH + c], h[i]);
}

__global__ void fc_out(const float* __restrict__ pool, const float* __restrict__ cnt,
                       const float* __restrict__ Wfc, const float* __restrict__ bfc,
                       float* __restrict__ out) {
  int i = blockIdx.x * blockDim.x + threadIdx.x;   // NGRAPH*NCLS = 8192
  if (i >= NGRAPH * NCLS) return;
  const int g = i / NCLS;
  const int j = i % NCLS;
  const float inv = 1.0f / fmaxf(cnt[g], 1.0f);
  float acc = bfc[j];
  #pragma unroll 8
  for (int c = 0; c < CH; ++c)
    acc += pool[g * CH + c] * inv * Wfc[c * NCLS + j];
  out[i] = acc;
}

// ---------------------------------------------------------------------------
// Host-side launcher
// ---------------------------------------------------------------------------
static inline size_t align256(size_t x) { return (x + 255) & ~(size_t)255; }

extern "C" void kernel_launch(void* const* d_in, const int* in_sizes, int n_in,
                              void* d_out, int out_size, void* d_ws, size_t ws_size,
                              hipStream_t stream) {
  const float* x    = (const float*)d_in[0];
  const int*   ei   = (const int*)  d_in[1];
  const int*   batch= (const int*)  d_in[2];
  const float* W1   = (const float*)d_in[3];
  const float* b1   = (const float*)d_in[4];
  const float* W2   = (const float*)d_in[5];
  const float* b2   = (const float*)d_in[6];
  const float* Wfc  = (const float*)d_in[7];
  const float* bfc  = (const float*)d_in[8];
  float* out = (float*)d_out;

  const int N = in_sizes[0] / CH;      // 100000
  const int E = in_sizes[1] / 2;       // 3200000
  const int* src = ei;
  const int* dst = ei + E;

  // Workspace carve-up
  char* ws = (char*)d_ws;
  size_t off = 0;
  float* xw   = (float*)(ws + off); off += align256((size_t)N * CH * 4);
  float* agg  = (float*)(ws + off); off += align256((size_t)N * CH * 4);
  float* deg  = (float*)(ws + off); off += align256((size_t)N * 4);
  float* dinv = (float*)(ws + off); off += align256((size_t)N * 4);
  float* pool = (float*)(ws + off); off += align256((size_t)NGRAPH * CH * 4);
  float* cnt  = (float*)(ws + off); off += align256((size_t)NGRAPH * 4);
  (void)ws_size; (void)n_in; (void)out_size;

  const int T = 256;
  const int gN    = (N + T - 1) / T;
  const int gE    = (E + T - 1) / T;
  const int gNC4  = (N * 32 + T - 1) / T;                 // N*32 float4
  const size_t NC = (size_t)N * CH;
  const int gNC   = (int)((NC + T - 1) / T);
  const int gTile = (N + 15) / 16;
  const int gEdge = (E + 7) / 8;                          // 8 edges (waves) / block

  // Degrees + symmetric normalization (shared by both conv layers)
  deg_init <<<gN, T, 0, stream>>>(deg, N);
  deg_count<<<gE, T, 0, stream>>>(dst, deg, E);
  deg_rsqrt<<<gN, T, 0, stream>>>(deg, dinv, N);

  // ---- Layer 1: relu(gcn_conv(x, W1, b1)) ----
  gemm_wmma_f32 <<<gTile, T, 0, stream>>>(x, W1, xw, N);
  self_loop_init<<<gNC4, T, 0, stream>>>(xw, dinv, agg, N);
  edge_scatter  <<<gEdge, T, 0, stream>>>(src, dst, dinv, xw, agg, E);
  bias_relu     <<<gNC, T, 0, stream>>>(agg, b1, NC);     // agg -> h1

  // ---- Layer 2: relu(gcn_conv(h1, W2, b2)) ----
  gemm_wmma_f32 <<<gTile, T, 0, stream>>>(agg, W2, xw, N);   // xw = h1 @ W2
  self_loop_init<<<gNC4, T, 0, stream>>>(xw, dinv, agg, N);  // h1 dead after GEMM
  edge_scatter  <<<gEdge, T, 0, stream>>>(src, dst, dinv, xw, agg, E);
  bias_relu     <<<gNC, T, 0, stream>>>(agg, b2, NC);     // agg -> h2

  // ---- Global mean pool + FC ----
  pool_zero <<<(NGRAPH * CH + T - 1) / T, T, 0, stream>>>(pool, cnt);
  pool_count<<<gN, T, 0, stream>>>(batch, cnt, N);
  pool_sum  <<<gNC, T, 0, stream>>>(agg, batch, pool, NC);
  fc_out    <<<(NGRAPH * NCLS + T - 1) / T, T, 0, stream>>>(pool, cnt, Wfc, bfc, out);
}